// VoxelDistill_41094247088580
// MI455X (gfx1250) — compile-verified
//
#include <hip/hip_runtime.h>
#include <math.h>

typedef float v2f __attribute__((ext_vector_type(2)));
typedef float v8f __attribute__((ext_vector_type(8)));

#define NBOX   100
#define FMAPSZ 128
#define EPS32  1.1920929e-07f

// workspace layout (floats):
// ws[0]                 : heatmap sum accumulator
// ws[8 .. 8+32768)      : heatmap, index = z*16384 + gx*128 + gy
// ws[8+32768 .. +12*b)  : per-box params, stride 12:
//   [0]=valid [1]=cxi [2]=cyi [3]=czi [4]=rx [5]=rz [6]=1/(2sx^2) [7]=1/(2sy^2) [8]=1/(2sz^2)
#define WS_HM   8
#define WS_BOX  (8 + 32768)

__device__ __forceinline__ float gauss_radius(float h, float w) {
    const float ov = 0.1f;
    float b1 = h + w;
    float c1 = w * h * (1.f - ov) / (1.f + ov);
    float r1 = (b1 + sqrtf(fmaxf(b1 * b1 - 4.f * c1, 0.f))) * 0.5f;
    float b2 = 2.f * (h + w);
    float c2 = (1.f - ov) * w * h;
    float r2 = (b2 + sqrtf(fmaxf(b2 * b2 - 16.f * c2, 0.f))) * 0.5f;
    float b3 = -2.f * ov * (h + w);
    float c3 = (ov - 1.f) * w * h;
    float r3 = (b3 + sqrtf(fmaxf(b3 * b3 - 16.f * ov * c3, 0.f))) * 0.5f;
    return fminf(fminf(r1, r2), r3);
}

__global__ void box_params_kernel(const float* __restrict__ boxes, float* __restrict__ ws) {
    int b = threadIdx.x;
    if (b == 0) ws[0] = 0.f;                 // reset sum accumulator each call
    if (b >= NBOX) return;
    const float* B = boxes + b * 9;
    float x = B[0], y = B[1], z = B[2];
    float wf = B[3] * 1.25f;                 // /VX/OSF = /0.8
    float lf = B[4] * 1.25f;
    float hf = B[5] * 0.625f;                // /VZ/OSF = /1.6
    float r_xy = gauss_radius(lf, wf);
    float r_z  = fmaxf(gauss_radius(lf, hf), gauss_radius(wf, hf));
    int rx = max(2, (int)truncf(r_xy * 2.5f));   // /0.4
    int rz = max(2, (int)truncf(r_z));           // /1.0
    float cx = (x + 51.2f) * 2.5f;
    float cy = (y + 51.2f) * 2.5f;
    float cz = (z + 5.0f);
    int cxi = (int)cx, cyi = (int)cy, czi = (int)cz;
    bool valid = (wf > 0.f) && (lf > 0.f) &&
                 (cxi >= 0) && (cxi < FMAPSZ) &&
                 (cyi >= 0) && (cyi < FMAPSZ) &&
                 (czi >= 0) && (czi <= 1);
    float sx = (2.f * (float)rx + 1.f) * (1.f / 6.f);
    float sz = (2.f * (float)rz + 1.f) * (1.f / 6.f);
    float* P = ws + WS_BOX + b * 12;
    P[0] = valid ? 1.f : 0.f;
    P[1] = (float)cxi;
    P[2] = (float)cyi;
    P[3] = (float)czi;
    P[4] = (float)rx;
    P[5] = (float)rz;
    P[6] = 1.f / (2.f * sx * sx);
    P[7] = 1.f / (2.f * sx * sx);
    P[8] = 1.f / (2.f * sz * sz);
}

__global__ void heatmap_kernel(const float* __restrict__ params, float* __restrict__ ws) {
    int idx = blockIdx.x * 256 + threadIdx.x;      // 0..32767 -> (z, gx, gy)
    int z  = idx >> 14;
    int r  = idx & 16383;
    int gx = r >> 7;
    int gy = r & 127;
    float m = 0.f;
    for (int b = 0; b < NBOX; ++b) {
        const float* P = params + b * 12;          // uniform -> scalar loads
        float dx = (float)gx - P[1];
        float dy = (float)gy - P[2];
        float dz = (float)z  - P[3];
        if ((P[0] != 0.f) && fabsf(dx) <= P[4] && fabsf(dy) <= P[4] && fabsf(dz) <= P[5]) {
            float g = expf(-(dx * dx * P[6] + dy * dy * P[7] + dz * dz * P[8]));
            if (g < EPS32) g = 0.f;
            m = fmaxf(m, g);
        }
    }
    ws[WS_HM + idx] = m;
    __shared__ float sm[256];
    sm[threadIdx.x] = m;
    __syncthreads();
    for (int s = 128; s > 0; s >>= 1) {
        if (threadIdx.x < s) sm[threadIdx.x] += sm[threadIdx.x + s];
        __syncthreads();
    }
    if (threadIdx.x == 0) atomicAdd(&ws[0], sm[0]);
}

// Fused: resize(student) -> 64x16 LDS tile -> 128x64 x 64x16 WMMA GEMM -> bias -> loss
// grid = 2048 blocks (one per 16-wide x-tile), block = 256 threads = 8 waves
__global__ void __launch_bounds__(256)
fused_gemm_loss_kernel(const float* __restrict__ student,
                       const float* __restrict__ teacher,
                       const float* __restrict__ convw,
                       const float* __restrict__ convb,
                       const float* __restrict__ ws,
                       float* __restrict__ out) {
    __shared__ float R[64][16];   // resized student tile: [in_channel][spatial]

    int t   = blockIdx.x;
    int zo  = t >> 10;            // 0..1
    int rem = t & 1023;
    int yo  = rem >> 3;           // 0..127
    int x0  = (rem & 7) << 4;     // 0,16,...,112
    int tid = threadIdx.x;
    int d0  = 1 + 4 * zo;         // depth lo sample: 1 or 5 (hi = d0+1, w = 0.5)

    // cooperative resize: 1024 values = 512 adjacent pairs; 2 pairs per thread.
    // Each pair consumes a contiguous, 16B-aligned 4-float span -> b128 loads.
#pragma unroll
    for (int i = 0; i < 2; ++i) {
        int idx = tid + 256 * i;   // 0..511
        int c   = idx >> 3;        // 0..63
        int p2  = idx & 7;         // pair index -> columns 2*p2, 2*p2+1
        const float4* p = (const float4*)(student + (size_t)c * 524288u
                                          + (size_t)d0 * 65536u
                                          + (size_t)(2 * yo) * 256u
                                          + (size_t)(2 * x0 + 4 * p2));
        float4 a0 = p[0];               // (d0,   2yo  )
        float4 a1 = p[64];              // (d0,   2yo+1)  (+256 floats)
        float4 a2 = p[16384];           // (d0+1, 2yo  )  (+65536 floats)
        float4 a3 = p[16384 + 64];      // (d0+1, 2yo+1)
        R[c][2 * p2 + 0] = 0.125f * (a0.x + a0.y + a1.x + a1.y + a2.x + a2.y + a3.x + a3.y);
        R[c][2 * p2 + 1] = 0.125f * (a0.z + a0.w + a1.z + a1.w + a2.z + a2.w + a3.z + a3.w);
    }
    __syncthreads();

    int lane  = tid & 31;
    int wv    = tid >> 5;
    int mbase = wv << 4;           // out-channel block per wave
    int mrow  = lane & 15;         // A: M index / B: N index (ISA layout)
    int khalf = lane >> 4;         // 0: K={0,1}, 1: K={2,3} within each K4 block

    v8f acc = {};
    const float* wr = convw + (size_t)(mbase + mrow) * 64 + 2 * khalf;
#pragma unroll
    for (int kb = 0; kb < 16; ++kb) {
        v2f a, b;
        a.x = wr[kb * 4 + 0];
        a.y = wr[kb * 4 + 1];
        int kr = kb * 4 + 2 * khalf;
        b.x = R[kr + 0][mrow];
        b.y = R[kr + 1][mrow];
        acc = __builtin_amdgcn_wmma_f32_16x16x4_f32(
            /*neg_a=*/false, a, /*neg_b=*/false, b,
            /*c_mod=*/(short)0, acc, /*reuse_a=*/false, /*reuse_b=*/false);
    }

    float scale = 10.0f / (ws[0] + 1e-4f);
    int n    = lane & 15;
    int xo   = x0 + n;
    int sidx = ((zo << 7) + yo) * 128 + xo;         // heatmap index (z,gx,gy)
    float hv = ws[WS_HM + sidx] * scale;

#pragma unroll
    for (int i = 0; i < 8; ++i) {
        int o = mbase + i + 8 * khalf;              // C/D layout: VGPR i -> M=i (+8 for hi half)
        float v = acc[i] + convb[o];
        size_t oidx = (size_t)((o * 2 + zo) * 128 + yo) * 128 + xo;
        out[oidx] = fabsf(v - teacher[oidx]) * hv;
    }
}

extern "C" void kernel_launch(void* const* d_in, const int* in_sizes, int n_in,
                              void* d_out, int out_size, void* d_ws, size_t ws_size,
                              hipStream_t stream) {
    const float* boxes   = (const float*)d_in[0];   // (100, 9)
    // d_in[1] = gt_labels_3d (unused by reference output)
    const float* teacher = (const float*)d_in[2];   // (1,128,2,128,128)
    const float* student = (const float*)d_in[3];   // (1,64,8,256,256)
    const float* convw   = (const float*)d_in[4];   // (128,64)
    const float* convb   = (const float*)d_in[5];   // (128,)
    float* out = (float*)d_out;                     // (1,128,2,128,128)
    float* ws  = (float*)d_ws;

    box_params_kernel<<<1, 128, 0, stream>>>(boxes, ws);
    heatmap_kernel<<<128, 256, 0, stream>>>(ws + WS_BOX, ws);
    fused_gemm_loss_kernel<<<2048, 256, 0, stream>>>(student, teacher, convw, convb, ws, out);
}